// SigLayer_5394478924084
// MI455X (gfx1250) — compile-verified
//
#include <hip/hip_runtime.h>

// Path signature, depth 4, d=6, L=100000.  Chen-identity chunked scan:
//   K1: per-chunk signature (LDS state, barrier-stepped Chen recursion)
//       + transposed copy of levels 1-3 for WMMA B operands
//   K2: sequential fold of levels 1-3 over chunks + transposed exclusive
//       prefixes R1,R2,R3 (WMMA A operands) + running sum of S4
//   K3: level-4 cross terms as 3 GEMMs over chunk axis (K=G) via
//       v_wmma_f32_16x16x4_f32, unconditional v2f loads (zero-page for pads)
//   K4: sum the three GEMM results into level-4 output
#define DCH  6
#define N4   1296
#define SIG  1554   // 6 + 36 + 216 + 1296
#define LOW  258    // 6 + 36 + 216
#define OFF2 6
#define OFF3 42
#define OFF4 258
#define G    256    // chunks (K dimension of the GEMMs)
#define LM1  99999  // number of increments

typedef __attribute__((ext_vector_type(2))) float v2f;
typedef __attribute__((ext_vector_type(8))) float v8f;

// ---------------- Kernel 1: per-chunk Chen recursion ----------------
__global__ __launch_bounds__(256) void sig_chunk_kernel(const float* __restrict__ x,
                                                        float* __restrict__ chunkSig,
                                                        float* __restrict__ SigLowT) {
  __shared__ float S[SIG];
  __shared__ float vsh[DCH];
  const int tid = threadIdx.x;
  const int g = blockIdx.x;
  const int start = (int)(((long long)g * LM1) / G);
  const int end   = (int)(((long long)(g + 1) * LM1) / G);

  for (int f = tid; f < SIG; f += 256) S[f] = 0.0f;
  __syncthreads();

  for (int i = start; i < end; ++i) {
    if (tid < DCH) vsh[tid] = x[(i + 1) * DCH + tid] - x[i * DCH + tid];
    __syncthreads();

    float nv[7];
#pragma unroll
    for (int e = 0; e < 7; ++e) {
      int f = tid + e * 256;
      if (f >= SIG) break;
      float t;
      if (f < OFF2) {                 // level 1
        t = S[f] + vsh[f];
      } else if (f < OFF3) {          // level 2: (a,b)
        int r = f - OFF2; int a = r / 6, b = r % 6;
        float va = vsh[a], vb = vsh[b];
        t = S[f] + S[a] * vb + va * vb * 0.5f;
      } else if (f < OFF4) {          // level 3: (a,b,c)
        int r = f - OFF3; int a = r / 36, b = (r / 6) % 6, c = r % 6;
        float va = vsh[a], vb = vsh[b], vc = vsh[c];
        t = S[f] + S[OFF2 + a * 6 + b] * vc
                 + S[a] * (vb * vc * 0.5f)
                 + va * vb * vc * (1.0f / 6.0f);
      } else {                        // level 4: (a,b,c,d)
        int r = f - OFF4; int a = r / 216, b = (r / 36) % 6, c = (r / 6) % 6, d = r % 6;
        float va = vsh[a], vb = vsh[b], vc = vsh[c], vd = vsh[d];
        t = S[f] + S[OFF3 + (a * 6 + b) * 6 + c] * vd
                 + S[OFF2 + a * 6 + b] * (vc * vd * 0.5f)
                 + S[a] * (vb * vc * vd * (1.0f / 6.0f))
                 + va * vb * vc * vd * (1.0f / 24.0f);
      }
      nv[e] = t;
    }
    __syncthreads();
#pragma unroll
    for (int e = 0; e < 7; ++e) {
      int f = tid + e * 256;
      if (f >= SIG) break;
      S[f] = nv[e];
    }
    __syncthreads();
  }

  for (int f = tid; f < SIG; f += 256) chunkSig[(long long)g * SIG + f] = S[f];
  // transposed low levels: SigLowT[f][g], contiguous over g (WMMA B operand)
  for (int f = tid; f < LOW; f += 256) SigLowT[(size_t)f * G + g] = S[f];
}

// ---- Kernel 2: fold levels 1-3 across chunks, record exclusive prefixes ----
__global__ __launch_bounds__(256) void sig_fold_kernel(const float* __restrict__ chunkSig,
                                                       float* __restrict__ RpreT,
                                                       float* __restrict__ zbuf,
                                                       float* __restrict__ out) {
  __shared__ float R[LOW];
  const int tid = threadIdx.x;
  zbuf[tid] = 0.0f;                          // zero page for padded WMMA lanes
  for (int f = tid; f < LOW; f += 256) R[f] = 0.0f;
  float s4[6];
#pragma unroll
  for (int e = 0; e < 6; ++e) s4[e] = 0.0f;
  __syncthreads();

  for (int g = 0; g < G; ++g) {
    const float* Sg = chunkSig + (long long)g * SIG;
    float nv[2];
#pragma unroll
    for (int e = 0; e < 2; ++e) {
      int f = tid + e * 256;
      if (f >= LOW) break;
      RpreT[(size_t)f * G + g] = R[f];       // exclusive prefix R_{g-1}, transposed
      float t;
      if (f < OFF2) {
        t = R[f] + Sg[f];
      } else if (f < OFF3) {
        int r = f - OFF2; int a = r / 6, b = r % 6;
        t = R[f] + Sg[f] + R[a] * Sg[b];
      } else {
        int r = f - OFF3; int a = r / 36, b = (r / 6) % 6, c = r % 6;
        t = R[f] + Sg[f] + R[a] * Sg[OFF2 + b * 6 + c]
                        + R[OFF2 + a * 6 + b] * Sg[c];
      }
      nv[e] = t;
    }
#pragma unroll
    for (int e = 0; e < 6; ++e) {            // running sum of S4_g
      int idx = tid + e * 256;
      if (idx < N4) s4[e] += Sg[OFF4 + idx];
    }
    __syncthreads();
#pragma unroll
    for (int e = 0; e < 2; ++e) {
      int f = tid + e * 256;
      if (f >= LOW) break;
      R[f] = nv[e];
    }
    __syncthreads();
  }

  for (int f = tid; f < LOW; f += 256) out[f] = R[f];
#pragma unroll
  for (int e = 0; e < 6; ++e) {
    int idx = tid + e * 256;
    if (idx < N4) out[OFF4 + idx] = s4[e];
  }
}

// ---- Kernel 3: level-4 cross terms as GEMMs over chunk axis (WMMA f32) ----
// T_c[abc][d]  = sum_g R3pre[g][abc] * S1_g[d]   (216 x 6  x K=G)
// T_b[ab][cd]  = sum_g R2pre[g][ab]  * S2_g[cd]  ( 36 x 36 x K=G)
// T_a[a][bcd]  = sum_g R1pre[g][a]   * S3_g[bcd] (  6 x 216x K=G)
// A[row][k] = RpreT[(aoff+row)*G + k], B[k][col] = SigLowT[(boff+col)*G + k]
__global__ __launch_bounds__(32) void sig_wmma_kernel(const float* __restrict__ RpreT,
                                                      const float* __restrict__ SigLowT,
                                                      const float* __restrict__ zbuf,
                                                      float* __restrict__ Tbuf) {
  const int b = blockIdx.x;
  int mdim, ndim, tileM, tileN, aoff, boff;
  float* outT;
  if (b < 14)      { mdim = 216; ndim = 6;   tileM = b;      tileN = 0;
                     aoff = OFF3; boff = 0;    outT = Tbuf; }
  else if (b < 23) { int t = b - 14; mdim = 36; ndim = 36; tileM = t / 3; tileN = t % 3;
                     aoff = OFF2; boff = OFF2; outT = Tbuf + N4; }
  else             { mdim = 6;   ndim = 216; tileM = 0;      tileN = b - 23;
                     aoff = 0;    boff = OFF3; outT = Tbuf + 2 * N4; }

  const int lane = threadIdx.x;
  const int half = lane >> 4;       // lane half selects K columns {0,1} vs {2,3}
  const int lidx = lane & 15;
  const int rowA = tileM * 16 + lidx;
  const int colB = tileN * 16 + lidx;
  // Branchless: out-of-range lanes read the zero page; loads are unconditional
  // so EXEC stays all-ones through the whole WMMA loop.
  const float* Aptr = (rowA < mdim) ? (RpreT   + (size_t)(aoff + rowA) * G) : zbuf;
  const float* Bptr = (colB < ndim) ? (SigLowT + (size_t)(boff + colB) * G) : zbuf;
  Aptr += 2 * half;                 // this lane-half supplies K = {ka, ka+1}
  Bptr += 2 * half;

  v8f acc = {};
  for (int g0 = 0; g0 < G; g0 += 4) {
    // A 16x4 f32 layout (ISA 7.12.2): VGPR0 = K{0,2}, VGPR1 = K{1,3};
    // K-adjacent values are contiguous in the transposed buffers -> one b64 each.
    v2f a  = *(const v2f*)(Aptr + g0);
    v2f bf = *(const v2f*)(Bptr + g0);
    acc = __builtin_amdgcn_wmma_f32_16x16x4_f32(false, a, false, bf,
                                                (short)0, acc, false, false);
  }

#pragma unroll
  for (int j = 0; j < 8; ++j) {
    // C/D layout: VGPR j -> M=j (lanes 0-15) / M=j+8 (lanes 16-31), N=lane%16
    int row = tileM * 16 + half * 8 + j;
    int col = tileN * 16 + lidx;
    if (row < mdim && col < ndim) outT[row * ndim + col] = acc[j];
  }
}

// ---------------- Kernel 4: add GEMM terms into level-4 output --------------
__global__ __launch_bounds__(256) void sig_final_kernel(const float* __restrict__ Tbuf,
                                                        float* __restrict__ out) {
  for (int i = threadIdx.x; i < N4; i += 256)
    out[OFF4 + i] += Tbuf[i] + Tbuf[N4 + i] + Tbuf[2 * N4 + i];
}

extern "C" void kernel_launch(void* const* d_in, const int* in_sizes, int n_in,
                              void* d_out, int out_size, void* d_ws, size_t ws_size,
                              hipStream_t stream) {
  (void)in_sizes; (void)n_in; (void)out_size; (void)ws_size;
  const float* x = (const float*)d_in[0];
  float* out = (float*)d_out;
  float* ws = (float*)d_ws;

  float* chunkSig = ws;                                // G*SIG floats
  float* SigLowT  = chunkSig + (size_t)G * SIG;        // LOW*G floats
  float* RpreT    = SigLowT  + (size_t)LOW * G;        // LOW*G floats
  float* zbuf     = RpreT    + (size_t)LOW * G;        // G floats (zero page)
  float* Tbuf     = zbuf     + (size_t)G;              // 3*N4 floats

  sig_chunk_kernel<<<G, 256, 0, stream>>>(x, chunkSig, SigLowT);
  sig_fold_kernel<<<1, 256, 0, stream>>>(chunkSig, RpreT, zbuf, out);
  sig_wmma_kernel<<<37, 32, 0, stream>>>(RpreT, SigLowT, zbuf, Tbuf);
  sig_final_kernel<<<1, 256, 0, stream>>>(Tbuf, out);
}